// AFNO2D_46308337386331
// MI455X (gfx1250) — compile-verified
//
#include <hip/hip_runtime.h>

// ---------------------------------------------------------------------------
// AFNO2D for MI455X (gfx1250): all stages as fp32 WMMA GEMMs
// (V_WMMA_F32_16X16X4_F32), DFT-as-GEMM with exact integer phase reduction.
// Shapes: B=4, H=90, W=180, C=768, NUM_BLOCKS=8, BLOCK=96, kept W-freqs=46.
// ---------------------------------------------------------------------------

typedef float v2f __attribute__((ext_vector_type(2)));
typedef float v8f __attribute__((ext_vector_type(8)));

__device__ __forceinline__ v8f zero8() {
  v8f z = {0.f, 0.f, 0.f, 0.f, 0.f, 0.f, 0.f, 0.f};
  return z;
}

__device__ __forceinline__ v8f wmma4(v2f a, v2f b, v8f c) {
  // D = A(16x4) * B(4x16) + C(16x16), all fp32
  return __builtin_amdgcn_wmma_f32_16x16x4_f32(
      /*neg_a=*/false, a, /*neg_b=*/false, b,
      /*c_mod=*/(short)0, c, /*reuse_a=*/false, /*reuse_b=*/false);
}

// A fragment (16x4 tile of row-major matrix, leading dim ld).
// ISA layout: lanes 0-15 hold (M=lane, K=0/1), lanes 16-31 hold (M=lane-16, K=2/3).
__device__ __forceinline__ v2f frag_a(const float* m, int ld) {
  int lane = threadIdx.x;
  const float* p = m + (long)(lane & 15) * ld + ((lane >> 4) << 1);
  v2f f; f.x = p[0]; f.y = p[1];
  return f;
}

// B fragment (4x16 tile, K rows x N cols, row-major, leading dim ld).
// Mirror of A: lanes 0-15 cover K=0/1, lanes 16-31 cover K=2/3 at N=lane&15.
__device__ __forceinline__ v2f frag_b(const float* m, long ld) {
  int lane = threadIdx.x;
  const float* p = m + (long)((lane >> 4) << 1) * ld + (lane & 15);
  v2f f; f.x = p[0]; f.y = p[ld];
  return f;
}

// C/D 16x16 layout: VGPR r, lanes 0-15 -> (M=r, N=lane); lanes 16-31 -> (M=r+8).
__device__ __forceinline__ void store_d_guard(float* out, long ld, v8f acc,
                                              int m0, int mmax) {
  int lane = threadIdx.x;
  int n = lane & 15;
  int mb = (lane >> 4) * 8;
#pragma unroll
  for (int r = 0; r < 8; ++r) {
    if (m0 + mb + r < mmax) out[(long)(mb + r) * ld + n] = acc[r];
  }
}

// ---------------------------------------------------------------------------
// K0: build padded DFT basis matrices in workspace (exact mod-phase).
//   CwF/SwF : 48x180  cos/sin(2*pi*w*q/180), rows w>=46 zero
//   P/Mm    : 96x92   s*(cos+sin)/(cos-sin)(2*pi*h*p/90), pads zero
//   ChI/ShI : 96x92   cos/sin(2*pi*p*h/90) (symmetric), pads zero
//   Cw2/NSw2: 192x48  s*cos / -s*sin (2*pi*q*w/180), pads zero
// ---------------------------------------------------------------------------
__global__ void k0_init(float* CwF, float* SwF, float* P, float* Mm,
                        float* ChI, float* ShI, float* Cw2, float* NSw2) {
  int i = blockIdx.x * blockDim.x + threadIdx.x;
  const float TWO_PI = 6.283185307179586f;
  const float s = 7.856742013e-3f;  // 1/sqrt(90*180)
  if (i < 48 * 180) {
    int w = i / 180, q = i % 180;
    float c = 0.f, sn = 0.f;
    if (w < 46) {
      int r = (w * q) % 180;
      sincosf(TWO_PI * (float)r / 180.f, &sn, &c);
    }
    CwF[i] = c;
    SwF[i] = sn;
  }
  if (i < 96 * 92) {
    int h = i / 92, p = i % 92;
    float c = 0.f, sn = 0.f;
    bool ok = (h < 90) && (p < 90);
    if (ok) {
      int r = (h * p) % 90;
      sincosf(TWO_PI * (float)r / 90.f, &sn, &c);
    }
    P[i] = ok ? s * (c + sn) : 0.f;
    Mm[i] = ok ? s * (c - sn) : 0.f;
    ChI[i] = ok ? c : 0.f;  // symmetric in (h,p): also valid as [p][h]
    ShI[i] = ok ? sn : 0.f;
  }
  if (i < 192 * 48) {
    int q = i / 48, w = i % 48;
    float c = 0.f, sn = 0.f;
    bool ok = (q < 180) && (w < 46);
    if (ok) {
      int r = (q * w) % 180;
      sincosf(TWO_PI * (float)r / 180.f, &sn, &c);
    }
    Cw2[i] = ok ? s * c : 0.f;
    NSw2[i] = ok ? -s * sn : 0.f;
  }
}

// ---------------------------------------------------------------------------
// K1: forward DFT along W. Per (b,h): Gr/Gi[w,c] = sum_q Cw/Sw[w,q]*x[q,c].
// Output layout Gr/Gi[b][w(46)][p=h(96 pad)][c], contiguous (p,c) per (b,w).
// grid = (B*H, 3 w-tiles, 24 c-pairs), block = 1 wave.
// ---------------------------------------------------------------------------
__global__ void k1_dftw(const float* __restrict__ x, const float* __restrict__ CwF,
                        const float* __restrict__ SwF, float* __restrict__ Gr,
                        float* __restrict__ Gi) {
  int bh = blockIdx.x;                 // b*90 + h
  int w0 = blockIdx.y * 16;
  int c0 = blockIdx.z * 32;
  int b = bh / 90, h = bh % 90;
  const float* xp = x + (long)bh * 180 * 768;  // X[q][c], ld=768
  v8f g0 = zero8(), g1 = zero8(), s0 = zero8(), s1 = zero8();
  for (int k = 0; k < 180; k += 4) {
    v2f aC = frag_a(CwF + (long)w0 * 180 + k, 180);
    v2f aS = frag_a(SwF + (long)w0 * 180 + k, 180);
    v2f b0 = frag_b(xp + (long)k * 768 + c0, 768);
    v2f b1 = frag_b(xp + (long)k * 768 + c0 + 16, 768);
    g0 = wmma4(aC, b0, g0);
    g1 = wmma4(aC, b1, g1);
    s0 = wmma4(aS, b0, s0);
    s1 = wmma4(aS, b1, s1);
  }
  long ldw = 96L * 768;  // stride between w rows in Gr/Gi
  float* grp = Gr + (((long)b * 46 + w0) * 96 + h) * 768 + c0;
  float* gip = Gi + (((long)b * 46 + w0) * 96 + h) * 768 + c0;
  store_d_guard(grp, ldw, g0, w0, 46);
  store_d_guard(grp + 16, ldw, g1, w0, 46);
  store_d_guard(gip, ldw, s0, w0, 46);
  store_d_guard(gip + 16, ldw, s1, w0, 46);
}

// ---------------------------------------------------------------------------
// K2: forward DFT along H + Re-Im combine + ortho norm.
// Per (b,w): Xh[h,c] = P[h,p]*Gr[p,c] + Mm[h,p]*Gi[p,c]  (K=92, zero-padded).
// Output S[b][w(46)][h(96)][c].  grid=(B*46, 6 h-tiles, 24 c-pairs).
// ---------------------------------------------------------------------------
__global__ void k2_dfth(const float* __restrict__ Gr, const float* __restrict__ Gi,
                        const float* __restrict__ P, const float* __restrict__ Mm,
                        float* __restrict__ S) {
  int bw = blockIdx.x;
  int h0 = blockIdx.y * 16;
  int c0 = blockIdx.z * 32;
  const float* grp = Gr + (long)bw * 96 * 768;
  const float* gip = Gi + (long)bw * 96 * 768;
  v8f a0 = zero8(), a1 = zero8();
  for (int k = 0; k < 92; k += 4) {
    v2f aP = frag_a(P + (long)h0 * 92 + k, 92);
    v2f aM = frag_a(Mm + (long)h0 * 92 + k, 92);
    v2f br0 = frag_b(grp + (long)k * 768 + c0, 768);
    v2f br1 = frag_b(grp + (long)k * 768 + c0 + 16, 768);
    v2f bi0 = frag_b(gip + (long)k * 768 + c0, 768);
    v2f bi1 = frag_b(gip + (long)k * 768 + c0 + 16, 768);
    a0 = wmma4(aP, br0, a0);
    a0 = wmma4(aM, bi0, a0);
    a1 = wmma4(aP, br1, a1);
    a1 = wmma4(aM, bi1, a1);
  }
  float* op = S + ((long)bw * 96 + h0) * 768 + c0;
  store_d_guard(op, 768, a0, h0, 90);
  store_d_guard(op + 16, 768, a1, h0, 90);
}

// ---------------------------------------------------------------------------
// K3: per-block MLP, in place on S.  Rows = B*46*96 = 17664 positions.
// Per (row-tile of 16, block n): o1 = relu(Xk@w1[0][n] + b1), routed through
// LDS to become the next A operand; o2 = o1@w2[0][n] + b2; softshrink.
// grid = (1104 row-tiles, 8 blocks), block = 1 wave.
// ---------------------------------------------------------------------------
__global__ void k3_mlp(float* __restrict__ S, const float* __restrict__ w1,
                       const float* __restrict__ b1, const float* __restrict__ w2,
                       const float* __restrict__ b2) {
  __shared__ float o1s[16][96];
  int rt = blockIdx.x;
  int n = blockIdx.y;
  int lane = threadIdx.x;
  float* base = S + (long)rt * 16 * 768 + n * 96;  // 16 x 96 tile, ld=768
  const float* W1 = w1 + (long)n * 96 * 96;        // w1[0][n][i][o]
  const float* W2 = w2 + (long)n * 96 * 96;
  const float* B1 = b1 + n * 96;
  const float* B2 = b2 + n * 96;
  int nn = lane & 15, mb = (lane >> 4) * 8;

  v8f acc[6];
#pragma unroll
  for (int t = 0; t < 6; ++t) acc[t] = zero8();
  for (int k = 0; k < 96; k += 4) {
    v2f a = frag_a(base + k, 768);
#pragma unroll
    for (int t = 0; t < 6; ++t) {
      v2f bb = frag_b(W1 + (long)k * 96 + t * 16, 96);
      acc[t] = wmma4(a, bb, acc[t]);
    }
  }
#pragma unroll
  for (int t = 0; t < 6; ++t) {
    float bv = B1[t * 16 + nn];
#pragma unroll
    for (int r = 0; r < 8; ++r) {
      float v = acc[t][r] + bv;
      o1s[mb + r][t * 16 + nn] = v > 0.f ? v : 0.f;  // relu
    }
  }
  __syncthreads();
#pragma unroll
  for (int t = 0; t < 6; ++t) acc[t] = zero8();
  for (int k = 0; k < 96; k += 4) {
    v2f a = frag_a(&o1s[0][0] + k, 96);
#pragma unroll
    for (int t = 0; t < 6; ++t) {
      v2f bb = frag_b(W2 + (long)k * 96 + t * 16, 96);
      acc[t] = wmma4(a, bb, acc[t]);
    }
  }
#pragma unroll
  for (int t = 0; t < 6; ++t) {
    float bv = B2[t * 16 + nn];
#pragma unroll
    for (int r = 0; r < 8; ++r) {
      float v = acc[t][r] + bv;
      // softshrink(lam=0.01)
      v = v > 0.01f ? v - 0.01f : (v < -0.01f ? v + 0.01f : 0.f);
      base[(long)(mb + r) * 768 + t * 16 + nn] = v;
    }
  }
}

// ---------------------------------------------------------------------------
// K4: inverse DFT along H. Per (b,w): U/V[p,c] = ChI/ShI[p,h]*S[h,c] (K=92).
// Output U/V[b][p(90)][w(48 pad)][c].  grid=(B*46, 6 p-tiles, 24 c-pairs).
// ---------------------------------------------------------------------------
__global__ void k4_idfth(const float* __restrict__ S, const float* __restrict__ ChI,
                         const float* __restrict__ ShI, float* __restrict__ U,
                         float* __restrict__ V) {
  int bw = blockIdx.x;
  int p0 = blockIdx.y * 16;
  int c0 = blockIdx.z * 32;
  int b = bw / 46, w = bw % 46;
  const float* sp = S + (long)bw * 96 * 768;
  v8f u0 = zero8(), u1 = zero8(), v0 = zero8(), v1 = zero8();
  for (int k = 0; k < 92; k += 4) {
    v2f aC = frag_a(ChI + (long)p0 * 92 + k, 92);
    v2f aS = frag_a(ShI + (long)p0 * 92 + k, 92);
    v2f b0 = frag_b(sp + (long)k * 768 + c0, 768);
    v2f b1 = frag_b(sp + (long)k * 768 + c0 + 16, 768);
    u0 = wmma4(aC, b0, u0);
    u1 = wmma4(aC, b1, u1);
    v0 = wmma4(aS, b0, v0);
    v1 = wmma4(aS, b1, v1);
  }
  long ldp = 48L * 768;
  float* up = U + (((long)b * 90 + p0) * 48 + w) * 768 + c0;
  float* vp = V + (((long)b * 90 + p0) * 48 + w) * 768 + c0;
  store_d_guard(up, ldp, u0, p0, 90);
  store_d_guard(up + 16, ldp, u1, p0, 90);
  store_d_guard(vp, ldp, v0, p0, 90);
  store_d_guard(vp + 16, ldp, v1, p0, 90);
}

// ---------------------------------------------------------------------------
// K5: inverse DFT along W + residual.  Per (b,p):
//   y[q,c] = sum_w Cw2[q,w]*U[w,c] + NSw2[q,w]*V[w,c] + x[q,c]   (K=48)
// Pad K-columns of Cw2/NSw2 are zero, so stale-but-finite U/V pad rows vanish.
// grid = (B*90, 12 q-tiles, 24 c-pairs).
// ---------------------------------------------------------------------------
__global__ void k5_idftw(const float* __restrict__ U, const float* __restrict__ V,
                         const float* __restrict__ Cw2, const float* __restrict__ NSw2,
                         const float* __restrict__ x, float* __restrict__ y) {
  int bp = blockIdx.x;  // b*90 + p
  int q0 = blockIdx.y * 16;
  int c0 = blockIdx.z * 32;
  const float* up = U + (long)bp * 48 * 768;
  const float* vp = V + (long)bp * 48 * 768;
  v8f a0 = zero8(), a1 = zero8();
  for (int k = 0; k < 48; k += 4) {
    v2f aC = frag_a(Cw2 + (long)q0 * 48 + k, 48);
    v2f aS = frag_a(NSw2 + (long)q0 * 48 + k, 48);
    v2f b0 = frag_b(up + (long)k * 768 + c0, 768);
    v2f b1 = frag_b(up + (long)k * 768 + c0 + 16, 768);
    v2f d0 = frag_b(vp + (long)k * 768 + c0, 768);
    v2f d1 = frag_b(vp + (long)k * 768 + c0 + 16, 768);
    a0 = wmma4(aC, b0, a0);
    a0 = wmma4(aS, d0, a0);
    a1 = wmma4(aC, b1, a1);
    a1 = wmma4(aS, d1, a1);
  }
  int lane = threadIdx.x;
  int nn = lane & 15, mb = (lane >> 4) * 8;
  const float* xp = x + ((long)bp * 180 + q0) * 768 + c0;
  float* yp = y + ((long)bp * 180 + q0) * 768 + c0;
#pragma unroll
  for (int r = 0; r < 8; ++r) {
    if (q0 + mb + r < 180) {
      long off = (long)(mb + r) * 768 + nn;
      yp[off] = a0[r] + xp[off];
      yp[off + 16] = a1[r] + xp[off + 16];
    }
  }
}

// ---------------------------------------------------------------------------
extern "C" void kernel_launch(void* const* d_in, const int* in_sizes, int n_in,
                              void* d_out, int out_size, void* d_ws, size_t ws_size,
                              hipStream_t stream) {
  (void)in_sizes; (void)n_in; (void)out_size; (void)ws_size;
  const float* x = (const float*)d_in[0];
  const float* w1 = (const float*)d_in[1];
  const float* b1 = (const float*)d_in[2];
  const float* w2 = (const float*)d_in[3];
  const float* b2 = (const float*)d_in[4];
  float* out = (float*)d_out;
  float* ws = (float*)d_ws;

  // Workspace carve-up (floats). Total = 40,768,896 floats = 163.1 MB.
  float* CwF = ws;                    // 48*180
  float* SwF = CwF + 48 * 180;        // 48*180
  float* P = SwF + 48 * 180;          // 96*92
  float* Mm = P + 96 * 92;            // 96*92
  float* ChI = Mm + 96 * 92;          // 96*92
  float* ShI = ChI + 96 * 92;         // 96*92
  float* Cw2 = ShI + 96 * 92;         // 192*48
  float* NSw2 = Cw2 + 192 * 48;       // 192*48
  float* Gr = NSw2 + 192 * 48;        // 4*46*96*768 = 13,565,952
  float* Gi = Gr + 13565952L;         // 13,565,952
  float* S = Gi + 13565952L;          // 13,565,952 (spectrum, in-place MLP)
  float* U = Gr;                      // alias: Gr dead after K2 (13,271,040 fits)
  float* V = Gi;                      // alias: Gi dead after K2

  k0_init<<<36, 256, 0, stream>>>(CwF, SwF, P, Mm, ChI, ShI, Cw2, NSw2);
  k1_dftw<<<dim3(4 * 90, 3, 24), 32, 0, stream>>>(x, CwF, SwF, Gr, Gi);
  k2_dfth<<<dim3(4 * 46, 6, 24), 32, 0, stream>>>(Gr, Gi, P, Mm, S);
  k3_mlp<<<dim3(1104, 8), 32, 0, stream>>>(S, w1, b1, w2, b2);
  k4_idfth<<<dim3(4 * 46, 6, 24), 32, 0, stream>>>(S, ChI, ShI, U, V);
  k5_idftw<<<dim3(4 * 90, 12, 24), 32, 0, stream>>>(U, V, Cw2, NSw2, x, out);
}